// DeformAtten2D_62526133895481
// MI455X (gfx1250) — compile-verified
//
#include <hip/hip_runtime.h>
#include <math.h>

// ---------------------------------------------------------------------------
// Deformable attention, MI455X (gfx1250):
//   - all GEMM stages on V_WMMA_F32_16X16X4_F32 (exact fp32 matrix-core path)
//   - attention K/V tiles staged to LDS via Tensor Data Mover (TENSOR_LOAD_TO_LDS),
//     double-buffered against the WMMA stream, synced with s_wait_tensorcnt.
// B=8, H=W=256, C=64.  wave32 throughout.
// ---------------------------------------------------------------------------

typedef __attribute__((ext_vector_type(2))) float v2f;
typedef __attribute__((ext_vector_type(8))) float v8f;
typedef __attribute__((ext_vector_type(4))) unsigned int v4u;
typedef __attribute__((ext_vector_type(4))) int v4i;
typedef __attribute__((ext_vector_type(8))) int v8i;

#define BB 8
#define HH 256
#define WW 256
#define CC 64
#define HW 65536L       // H*W
#define NPIX 524288L    // B*H*W

static __device__ __forceinline__ v8f vzero8() {
  v8f z = {0.f, 0.f, 0.f, 0.f, 0.f, 0.f, 0.f, 0.f};
  return z;
}

static __device__ __forceinline__ v8f wmma4(v2f a, v2f b, v8f c) {
  // D = A(16x4 f32) * B(4x16 f32) + C(16x16 f32)
  return __builtin_amdgcn_wmma_f32_16x16x4_f32(false, a, false, b, (short)0, c,
                                               false, false);
}

// Issue one 2D TDM tile load: global (tile_d1 rows x tile_d0 f32, row stride
// 256 f32) -> contiguous LDS at lds_addr.  Wave-level async op (TENSORcnt).
static __device__ __forceinline__ void tdm_load_2d(unsigned lds_addr,
                                                   unsigned long long gaddr,
                                                   unsigned tile_d0,
                                                   unsigned tile_d1) {
  v4u g0;
  g0.x = 1u;                                   // count=1, user mode, no gather
  g0.y = lds_addr;                             // LDS byte address
  g0.z = (unsigned)(gaddr & 0xffffffffULL);    // global addr [31:0]
  g0.w = (unsigned)((gaddr >> 32) & 0x01ffffffULL) | (2u << 30);  // [56:32]|type=2
  v8i g1;
  g1[0] = 0x00020000;                          // workgroup_mask=0, data_size=4B
  g1[1] = (int)(256u << 16);                   // tensor_dim0=256 (low16 @bit48)
  g1[2] = (int)(256u << 16);                   // tensor_dim0 hi | tensor_dim1=256
  g1[3] = (int)(tile_d0 << 16);                // tensor_dim1 hi | tile_dim0
  g1[4] = (int)tile_d1;                        // tile_dim1 | tile_dim2=0
  g1[5] = 256;                                 // tensor_dim0_stride = 256 f32
  g1[6] = 0;
  g1[7] = 0;
  v4i gz4 = {0, 0, 0, 0};
  v8i gz8 = {0, 0, 0, 0, 0, 0, 0, 0};
  // 6-arg toolchain signature: (g0, g1, g2, g3, g4, cpol)
  __builtin_amdgcn_tensor_load_to_lds(g0, g1, gz4, gz4, gz8, 0);
}

// ---------------------------------------------------------------------------
// Kernel 1: Q = x @ Wq^T + bq.  One wave per 16-pixel tile.  Writes q in
// channel-planar layout (attention heads) AND pixel-major layout (conv GEMM).
// ---------------------------------------------------------------------------
__global__ __launch_bounds__(32)
void k_qproj(const float* __restrict__ x, const float* __restrict__ Wq,
             const float* __restrict__ bq, float* __restrict__ q_chw,
             float* __restrict__ q_pix) {
  const int lane = threadIdx.x;
  const int hi = lane >> 4;
  const int ln = lane & 15;
  const long n0 = (long)blockIdx.x * 16;
  const float* xrow = x + (n0 + ln) * 64;  // A row for this lane (M = ln)

  v8f acc[4];
#pragma unroll
  for (int t = 0; t < 4; ++t) acc[t] = vzero8();

  for (int k0 = 0; k0 < 64; k0 += 4) {
    const int ka = k0 + 2 * hi;
    v2f a;
    a.x = xrow[ka];
    a.y = xrow[ka + 1];
#pragma unroll
    for (int t = 0; t < 4; ++t) {
      const int oc = t * 16 + ln;  // B column
      v2f bf;
      bf.x = Wq[oc * 64 + ka];     // B[k,oc] = Wq[oc,k]
      bf.y = Wq[oc * 64 + ka + 1];
      acc[t] = wmma4(a, bf, acc[t]);
    }
  }

  const int b = (int)(n0 >> 16);
  const int hw = (int)(n0 & 65535);
#pragma unroll
  for (int t = 0; t < 4; ++t) {
    const int oc = t * 16 + ln;
    const float bias = bq[oc];
    float* qc = q_chw + ((long)(b * 64 + oc) << 16) + hw;
#pragma unroll
    for (int r = 0; r < 8; ++r) {
      const int m = r + 8 * hi;          // row within tile
      const float v = acc[t][r] + bias;
      qc[m] = v;                         // [B,C,H,W]
      q_pix[(n0 + m) * 64 + oc] = v;     // [N,64]
    }
  }
}

// ---------------------------------------------------------------------------
// Kernel 2: 5x5 conv over Q (implicit GEMM, 25 shifted taps), fused with the
// 64->2 offset projection + tanh.  2 waves per block, 32-pixel tile in a row.
// LDS: 5x36x64 Q neighborhood (46KB) + per-tap 64x64 weight slice (16KB).
// ---------------------------------------------------------------------------
__global__ __launch_bounds__(64)
void k_offsets(const float* __restrict__ q_pix, const float* __restrict__ Woff1,
               const float* __restrict__ boff1, const float* __restrict__ Woff2,
               float* __restrict__ offb) {
  __shared__ float nb[5][36][64];
  __shared__ float wt[64 * 64];

  const int tid = threadIdx.x;
  const int lane = tid & 31;
  const int wave = tid >> 5;
  const int hi = (lane >> 4) & 1;
  const int ln = lane & 15;

  const int blk = blockIdx.x;      // B*H*(W/32) blocks
  const int wb = blk & 7;
  const int h = (blk >> 3) & 255;
  const int b = blk >> 11;
  const int w0 = wb * 32;

  // Stage Q neighborhood rows h-2..h+2, cols w0-2..w0+33 (zero padded).
  for (int idx = tid; idx < 5 * 36; idx += 64) {
    const int dy = idx / 36;
    const int j = idx % 36;
    const int hs = h + dy - 2;
    const int ws = w0 + j - 2;
    float* dst = &nb[dy][j][0];
    if (hs >= 0 && hs < HH && ws >= 0 && ws < WW) {
      const float* src = q_pix + (((long)b << 16) + hs * 256 + ws) * 64;
      for (int c = 0; c < 64; ++c) dst[c] = src[c];
    } else {
      for (int c = 0; c < 64; ++c) dst[c] = 0.f;
    }
  }
  __syncthreads();

  v8f acc[4];
#pragma unroll
  for (int t = 0; t < 4; ++t) acc[t] = vzero8();
  const int pxl = wave * 16;  // this wave's local pixel base

  for (int tap = 0; tap < 25; ++tap) {
    const int dy = tap / 5;
    const int dx = tap % 5;
    __syncthreads();
    // wt[ic*64+oc] = Woff1[oc, ic, dy, dx]
    for (int idx = tid; idx < 4096; idx += 64) {
      const int ic = idx >> 6, oc = idx & 63;
      wt[idx] = Woff1[(oc * 64 + ic) * 25 + tap];
    }
    __syncthreads();

    for (int k0 = 0; k0 < 64; k0 += 4) {
      const int ka = k0 + 2 * hi;
      v2f a;
      a.x = nb[dy][pxl + ln + dx][ka];
      a.y = nb[dy][pxl + ln + dx][ka + 1];
#pragma unroll
      for (int t = 0; t < 4; ++t) {
        const int oc = t * 16 + ln;
        v2f bf;
        bf.x = wt[ka * 64 + oc];
        bf.y = wt[(ka + 1) * 64 + oc];
        acc[t] = wmma4(a, bf, acc[t]);
      }
    }
  }

  // Fused: o += boff1; offset_j = tanh(sum_oc o*Woff2[j,oc]) * 5
  float p0[8], p1[8];
#pragma unroll
  for (int r = 0; r < 8; ++r) { p0[r] = 0.f; p1[r] = 0.f; }
#pragma unroll
  for (int t = 0; t < 4; ++t) {
    const int oc = t * 16 + ln;
    const float bo = boff1[oc];
    const float w0c = Woff2[oc];
    const float w1c = Woff2[64 + oc];
#pragma unroll
    for (int r = 0; r < 8; ++r) {
      const float o = acc[t][r] + bo;
      p0[r] += o * w0c;
      p1[r] += o * w1c;
    }
  }
  // Reduce across the 16 lanes of each half-wave (cols); rows live in r + 8*hi.
#pragma unroll
  for (int m = 1; m < 16; m <<= 1) {
#pragma unroll
    for (int r = 0; r < 8; ++r) {
      p0[r] += __shfl_xor(p0[r], m, 32);
      p1[r] += __shfl_xor(p1[r], m, 32);
    }
  }
  if (ln == 0) {
    for (int r = 0; r < 8; ++r) {
      const int m = r + 8 * hi;
      const long n = ((long)b << 16) + h * 256 + (w0 + pxl + m);
      offb[n * 2 + 0] = tanhf(p0[r]) * 5.f;
      offb[n * 2 + 1] = tanhf(p1[r]) * 5.f;
    }
  }
}

// ---------------------------------------------------------------------------
// Kernel 3: bilinear grid-sample of x (channel rows contiguous in pixel-major
// input) fused with K/V 1x1 projections (+bias, +rpb).  One wave / 16 pixels.
// ---------------------------------------------------------------------------
__global__ __launch_bounds__(32)
void k_sample_kv(const float* __restrict__ x, const float* __restrict__ offb,
                 const float* __restrict__ Wk, const float* __restrict__ bk,
                 const float* __restrict__ Wv, const float* __restrict__ bv,
                 const float* __restrict__ rpb, float* __restrict__ k_chw,
                 float* __restrict__ v_chw) {
  __shared__ float xs[16][64];
  const int lane = threadIdx.x;
  const int hi = lane >> 4;
  const int ln = lane & 15;

  const int blk = blockIdx.x;  // B*H*(W/16)
  const int wb = blk & 15;
  const int h = (blk >> 4) & 255;
  const int b = blk >> 12;
  const int w0 = wb * 16;
  const float* xb = x + ((long)b << 16) * 64;

  for (int p = 0; p < 16; ++p) {
    const long n = ((long)b << 16) + h * 256 + (w0 + p);
    const float vgx = (float)(w0 + p) + offb[n * 2 + 0];
    const float vgy = (float)h + offb[n * 2 + 1];
    const float gx = 2.f * vgx / 255.f - 1.f;
    const float gy = 2.f * vgy / 255.f - 1.f;
    const float ix = ((gx + 1.f) * 256.f - 1.f) * 0.5f;
    const float iy = ((gy + 1.f) * 256.f - 1.f) * 0.5f;
    const float ix0f = floorf(ix), iy0f = floorf(iy);
    const int ix0 = (int)ix0f, iy0 = (int)iy0f;
    const float wx1 = ix - ix0f, wy1 = iy - iy0f;
    const float wx0 = 1.f - wx1, wy0 = 1.f - wy1;

    const int c = lane * 2;  // each lane blends 2 channels
    float a0 = 0.f, a1 = 0.f;
#pragma unroll
    for (int cy = 0; cy < 2; ++cy) {
      const int yi = iy0 + cy;
      const float wy = cy ? wy1 : wy0;
      if (yi >= 0 && yi < HH) {
#pragma unroll
        for (int cx = 0; cx < 2; ++cx) {
          const int xi = ix0 + cx;
          const float wgt = wy * (cx ? wx1 : wx0);
          if (xi >= 0 && xi < WW) {
            const float* src = xb + ((long)(yi * 256 + xi)) * 64 + c;
            a0 += wgt * src[0];
            a1 += wgt * src[1];
          }
        }
      }
    }
    xs[p][c] = a0;
    xs[p][c + 1] = a1;
  }
  __syncthreads();

  v8f ak[4], av[4];
#pragma unroll
  for (int t = 0; t < 4; ++t) { ak[t] = vzero8(); av[t] = vzero8(); }

  for (int k0 = 0; k0 < 64; k0 += 4) {
    const int ka = k0 + 2 * hi;
    v2f a;
    a.x = xs[ln][ka];
    a.y = xs[ln][ka + 1];
#pragma unroll
    for (int t = 0; t < 4; ++t) {
      const int oc = t * 16 + ln;
      v2f bkf, bvf;
      bkf.x = Wk[oc * 64 + ka];
      bkf.y = Wk[oc * 64 + ka + 1];
      ak[t] = wmma4(a, bkf, ak[t]);
      bvf.x = Wv[oc * 64 + ka];
      bvf.y = Wv[oc * 64 + ka + 1];
      av[t] = wmma4(a, bvf, av[t]);
    }
  }

#pragma unroll
  for (int t = 0; t < 4; ++t) {
    const int oc = t * 16 + ln;
    const float bkb = bk[oc];
    const float bvb = bv[oc] + rpb[oc * 256 + h];  // rpb[1,C,H,1]
    const long base = (((long)(b * 64 + oc)) << 16) + h * 256 + w0;
#pragma unroll
    for (int r = 0; r < 8; ++r) {
      const int m = r + 8 * hi;
      k_chw[base + m] = ak[t][r] + bkb;
      v_chw[base + m] = av[t][r] + bvb;
    }
  }
}

// ---------------------------------------------------------------------------
// Kernel 4: attention per (head, 16-row tile).  K tiles (16x256) and V tiles
// (256x16) are DMA'd into LDS by the Tensor Data Mover, double-buffered
// against the WMMA stream (TENSORcnt).  Scores tile lives in LDS; row softmax
// via wave reductions + v_exp_f32.  The 134MB score tensor never reaches HBM.
// ---------------------------------------------------------------------------
__global__ __launch_bounds__(32)
void k_attn(const float* __restrict__ q_chw, const float* __restrict__ k_chw,
            const float* __restrict__ v_chw, float* __restrict__ aout) {
  __shared__ float s[16][256];     // 16KB score tile
  __shared__ float tb[2][4096];    // 2 x 16KB TDM staging buffers
  const int lane = threadIdx.x;
  const int hi = lane >> 4;
  const int ln = lane & 15;
  const long head = blockIdx.y;
  const int i0 = blockIdx.x * 16;
  const float* Q = q_chw + (head << 16);
  const float* K = k_chw + (head << 16);
  const float* V = v_chw + (head << 16);

  const unsigned lds_tb = (unsigned)(size_t)(&tb[0][0]);
  const unsigned long long Kb = (unsigned long long)(size_t)K;
  const unsigned long long Vb = (unsigned long long)(size_t)V;

  // ---- phase 1: scores[i,j] = (1/8) * sum_d Q[i,d] K[j,d] --------------
  tdm_load_2d(lds_tb, Kb, 256u, 16u);  // K rows 0..15
  for (int jt = 0; jt < 16; ++jt) {
    if (jt < 15) {
      tdm_load_2d(lds_tb + (unsigned)(((jt + 1) & 1) * 16384),
                  Kb + (unsigned long long)(jt + 1) * 16384ULL, 256u, 16u);
      __builtin_amdgcn_s_wait_tensorcnt(1);  // tile jt landed (in-order)
    } else {
      __builtin_amdgcn_s_wait_tensorcnt(0);
    }
    const float* kb = &tb[jt & 1][0];  // [16 rows][256]
    v8f acc = vzero8();
    for (int k0 = 0; k0 < 256; k0 += 4) {
      const int ka = k0 + 2 * hi;
      v2f a, bf;
      a.x = Q[(i0 + ln) * 256 + ka];
      a.y = Q[(i0 + ln) * 256 + ka + 1];
      bf.x = kb[ln * 256 + ka];          // B[k,j] = K[j,k]
      bf.y = kb[ln * 256 + ka + 1];
      acc = wmma4(a, bf, acc);
    }
#pragma unroll
    for (int r = 0; r < 8; ++r) s[r + 8 * hi][jt * 16 + ln] = acc[r] * 0.125f;
  }
  __syncthreads();

  // ---- row softmax ------------------------------------------------------
  for (int r = 0; r < 16; ++r) {
    float m = -3.402823e38f;
    for (int t = lane; t < 256; t += 32) m = fmaxf(m, s[r][t]);
#pragma unroll
    for (int o = 16; o > 0; o >>= 1) m = fmaxf(m, __shfl_xor(m, o, 32));
    float sum = 0.f;
    for (int t = lane; t < 256; t += 32) {
      const float e = __expf(s[r][t] - m);
      s[r][t] = e;
      sum += e;
    }
#pragma unroll
    for (int o = 16; o > 0; o >>= 1) sum += __shfl_xor(sum, o, 32);
    const float inv = 1.f / sum;
    for (int t = lane; t < 256; t += 32) s[r][t] *= inv;
  }
  __syncthreads();

  // ---- phase 2: out[i,d] = sum_j P[i,j] V[j,d] --------------------------
  tdm_load_2d(lds_tb, Vb, 16u, 256u);  // V cols 0..15 (256 rows x 16 f32)
  for (int dt = 0; dt < 16; ++dt) {
    if (dt < 15) {
      tdm_load_2d(lds_tb + (unsigned)(((dt + 1) & 1) * 16384),
                  Vb + (unsigned long long)(dt + 1) * 64ULL, 16u, 256u);
      __builtin_amdgcn_s_wait_tensorcnt(1);
    } else {
      __builtin_amdgcn_s_wait_tensorcnt(0);
    }
    const float* vb = &tb[dt & 1][0];  // [256 rows][16]
    v8f acc = vzero8();
    for (int k0 = 0; k0 < 256; k0 += 4) {
      const int ka = k0 + 2 * hi;
      v2f a, bf;
      a.x = s[ln][ka];
      a.y = s[ln][ka + 1];
      bf.x = vb[ka * 16 + ln];
      bf.y = vb[(ka + 1) * 16 + ln];
      acc = wmma4(a, bf, acc);
    }
    const long base = (head << 16) + (long)i0 * 256 + dt * 16;
#pragma unroll
    for (int r = 0; r < 8; ++r)
      aout[base + (long)(r + 8 * hi) * 256 + ln] = acc[r];
  }
}

// ---------------------------------------------------------------------------
// Kernel 5: out = reshape(aout) @ Wout^T + bout.  The reference's raw reshape
// means aout's flat buffer IS pixel-major [N,64]; output is [B,H,W,C].
// ---------------------------------------------------------------------------
__global__ __launch_bounds__(32)
void k_outproj(const float* __restrict__ ain, const float* __restrict__ Wout,
               const float* __restrict__ bout, float* __restrict__ out) {
  const int lane = threadIdx.x;
  const int hi = lane >> 4;
  const int ln = lane & 15;
  const long n0 = (long)blockIdx.x * 16;
  const float* arow = ain + (n0 + ln) * 64;

  v8f acc[4];
#pragma unroll
  for (int t = 0; t < 4; ++t) acc[t] = vzero8();

  for (int k0 = 0; k0 < 64; k0 += 4) {
    const int ka = k0 + 2 * hi;
    v2f a;
    a.x = arow[ka];
    a.y = arow[ka + 1];
#pragma unroll
    for (int t = 0; t < 4; ++t) {
      const int oc = t * 16 + ln;
      v2f bf;
      bf.x = Wout[oc * 64 + ka];
      bf.y = Wout[oc * 64 + ka + 1];
      acc[t] = wmma4(a, bf, acc[t]);
    }
  }
#pragma unroll
  for (int t = 0; t < 4; ++t) {
    const int oc = t * 16 + ln;
    const float bb = bout[oc];
#pragma unroll
    for (int r = 0; r < 8; ++r)
      out[(n0 + r + 8 * hi) * 64 + oc] = acc[t][r] + bb;
  }
}

// ---------------------------------------------------------------------------
extern "C" void kernel_launch(void* const* d_in, const int* in_sizes, int n_in,
                              void* d_out, int out_size, void* d_ws,
                              size_t ws_size, hipStream_t stream) {
  const float* x = (const float*)d_in[0];
  const float* Wq = (const float*)d_in[1];
  const float* bq = (const float*)d_in[2];
  const float* Wk = (const float*)d_in[3];
  const float* bk = (const float*)d_in[4];
  const float* Wv = (const float*)d_in[5];
  const float* bv = (const float*)d_in[6];
  const float* Woff1 = (const float*)d_in[7];
  const float* boff1 = (const float*)d_in[8];
  const float* Woff2 = (const float*)d_in[9];
  const float* rpb = (const float*)d_in[10];
  const float* Wout = (const float*)d_in[11];
  const float* bout = (const float*)d_in[12];

  float* ws = (float*)d_ws;
  float* q_chw = ws;                         // 128MB
  float* q_pix = q_chw + (NPIX * 64);        // 128MB
  float* offb = q_pix + (NPIX * 64);         // 4MB
  float* k_chw = offb + (NPIX * 2);          // 128MB
  float* v_chw = k_chw + (NPIX * 64);        // 128MB
  float* aout = v_chw + (NPIX * 64);         // 128MB

  k_qproj<<<(int)(NPIX / 16), 32, 0, stream>>>(x, Wq, bq, q_chw, q_pix);
  k_offsets<<<BB * HH * (WW / 32), 64, 0, stream>>>(q_pix, Woff1, boff1, Woff2,
                                                    offb);
  k_sample_kv<<<BB * HH * (WW / 16), 32, 0, stream>>>(x, offb, Wk, bk, Wv, bv,
                                                      rpb, k_chw, v_chw);
  dim3 ga(16, 512);
  k_attn<<<ga, 32, 0, stream>>>(q_chw, k_chw, v_chw, aout);
  k_outproj<<<(int)(NPIX / 16), 32, 0, stream>>>(aout, Wout, bout,
                                                 (float*)d_out);
}